// ContrastiveLoss_56298431316316
// MI455X (gfx1250) — compile-verified
//
#include <hip/hip_runtime.h>
#include <hip/hip_bf16.h>
#include <math.h>

typedef __attribute__((ext_vector_type(2))) float v2f;
typedef __attribute__((ext_vector_type(8))) float v8f;

#define B1N 8192
#define KD 128
#define PITCH 132           // 128 + 4 dword padding -> conflict-free ds_load_b64 fragments
#define BT 128              // block tile (both dims)
#define TEMP_INV (1.0f / 0.07f)
#define EPS_F 1e-8f

// ---------------------------------------------------------------- init / final
__global__ void cl_init_kernel(double* acc) {
    if (threadIdx.x == 0) { acc[0] = 0.0; acc[1] = 0.0; }
}

__global__ void cl_final_kernel(const double* acc, float* out) {
    if (threadIdx.x == 0) {
        // -log(loss1/loss2) = log(loss2) - log(loss1)
        out[0] = (float)(log(acc[1]) - log(acc[0]));
    }
}

// ---------------------------------------------------------------- row norms
// One wave per row; 16384 rows total (n1 then n2).
__global__ __launch_bounds__(256)
void cl_norms_kernel(const float* __restrict__ e1, const float* __restrict__ e2,
                     float* __restrict__ nrm) {
    const int lane = threadIdx.x & 31;
    const int wid  = threadIdx.x >> 5;
    const int row  = blockIdx.x * 8 + wid;          // 0..16383
    const float* src = (row < B1N) ? (e1 + (size_t)row * KD)
                                   : (e2 + (size_t)(row - B1N) * KD);
    float4 q = ((const float4*)src)[lane];          // 32 lanes x 4 floats = 128
    float ss = q.x * q.x + q.y * q.y + q.z * q.z + q.w * q.w;
#pragma unroll
    for (int o = 16; o > 0; o >>= 1) ss += __shfl_xor(ss, o, 32);
    if (lane == 0) nrm[row] = sqrtf(ss);
}

// ---------------------------------------------------------------- fused Gram + exp + reduce
// blockIdx.z == 0: sim11 with mask ((col - i) mod B1) > i ; z == 1: sim12, no mask.
// 8 waves tile the 128x128 block as 4 row-groups x 2 col-groups;
// each wave computes 32 rows x 64 cols = 2x4 WMMA tiles.
__global__ __launch_bounds__(256)
void cl_gram_exp_kernel(const float* __restrict__ e1, const float* __restrict__ e2,
                        const float* __restrict__ nrm, double* __restrict__ accum) {
    __shared__ float lA[BT * PITCH];
    __shared__ float lB[BT * PITCH];
    __shared__ float red[8];

    const int tid  = threadIdx.x;
    const int lane = tid & 31;
    const int wid  = tid >> 5;
    const int z    = blockIdx.z;
    const int i0   = blockIdx.x * BT;
    const int j0   = blockIdx.y * BT;

    const float* Bsrc = (z == 0) ? e1 : e2;
    const float* nrmB = nrm + ((z == 0) ? 0 : B1N);

    // Cooperative load: each iteration, one wave writes one full padded LDS row.
#pragma unroll
    for (int it = 0; it < 16; ++it) {
        int f4  = it * 256 + tid;                 // 0..4095 float4 index
        int row = f4 >> 5;                        // 0..127
        int c4  = (f4 & 31) << 2;                 // 0,4,...,124
        float4 qa = ((const float4*)(e1   + (size_t)(i0 + row) * KD))[f4 & 31];
        float4 qb = ((const float4*)(Bsrc + (size_t)(j0 + row) * KD))[f4 & 31];
        *(float4*)&lA[row * PITCH + c4] = qa;
        *(float4*)&lB[row * PITCH + c4] = qb;
    }
    __syncthreads();

    // WMMA f32 16x16x4 fragment addressing (wave32):
    //   A (16x4): lane l<16 holds A[l, k..k+1], lane l+16 holds A[l, k+2..k+3]
    //   B (4x16): lane l<16 holds B[k..k+1, l], lane l+16 holds B[k+2..k+3, l]
    // Both are contiguous float2 at row (lane&15), column k + 2*(lane>>4).
    const int rsel = lane & 15;
    const int koff = (lane >> 4) << 1;

    const int wy = wid & 3;     // row group: 32 rows each
    const int wx = wid >> 2;    // col group: 64 cols each

    v8f acc[2][4] = {};
    const float* aBase = &lA[(wy * 32 + rsel) * PITCH + koff];
    const float* bBase = &lB[(wx * 64 + rsel) * PITCH + koff];

    for (int k0 = 0; k0 < KD; k0 += 4) {
        v2f a0 = *(const v2f*)(aBase + k0);
        v2f a1 = *(const v2f*)(aBase + 16 * PITCH + k0);
#pragma unroll
        for (int t = 0; t < 4; ++t) {
            v2f b = *(const v2f*)(bBase + t * 16 * PITCH + k0);
            acc[0][t] = __builtin_amdgcn_wmma_f32_16x16x4_f32(
                false, a0, false, b, (short)0, acc[0][t], false, false);
            acc[1][t] = __builtin_amdgcn_wmma_f32_16x16x4_f32(
                false, a1, false, b, (short)0, acc[1][t], false, false);
        }
    }

    // Epilogue: C layout vgpr r -> M = r + 8*(lane>=16), N = lane&15.
    const int hi = lane >> 4;
    float niv[2][8];
#pragma unroll
    for (int s = 0; s < 2; ++s)
#pragma unroll
        for (int r = 0; r < 8; ++r)
            niv[s][r] = nrm[i0 + wy * 32 + s * 16 + r + 8 * hi];

    float psum = 0.0f;
#pragma unroll
    for (int t = 0; t < 4; ++t) {
        int col  = j0 + wx * 64 + t * 16 + rsel;
        float nj = nrmB[col];
#pragma unroll
        for (int s = 0; s < 2; ++s) {
#pragma unroll
            for (int r = 0; r < 8; ++r) {
                int i     = i0 + wy * 32 + s * 16 + r + 8 * hi;
                float sim = acc[s][t][r] / fmaxf(niv[s][r] * nj, EPS_F);
                float ev  = __expf(sim * TEMP_INV);
                bool inc  = true;
                if (z == 0) {
                    int d = col - i;
                    d += (d >> 31) & B1N;         // (col - i) mod B1
                    inc = d > i;
                }
                psum += inc ? ev : 0.0f;
            }
        }
    }

#pragma unroll
    for (int o = 16; o > 0; o >>= 1) psum += __shfl_xor(psum, o, 32);
    if (lane == 0) red[wid] = psum;
    __syncthreads();
    if (tid == 0) {
        float s = 0.0f;
#pragma unroll
        for (int w = 0; w < 8; ++w) s += red[w];
        atomicAdd(&accum[z], (double)s);
    }
}

// ---------------------------------------------------------------- launch
extern "C" void kernel_launch(void* const* d_in, const int* in_sizes, int n_in,
                              void* d_out, int out_size, void* d_ws, size_t ws_size,
                              hipStream_t stream) {
    const float* e1 = (const float*)d_in[0];
    const float* e2 = (const float*)d_in[1];
    float* out = (float*)d_out;

    // workspace: [0,16) two f64 accumulators; [16, 16+64KB) row norms (n1 then n2)
    double* accum = (double*)d_ws;
    float*  nrm   = (float*)((char*)d_ws + 16);

    cl_init_kernel<<<1, 32, 0, stream>>>(accum);
    cl_norms_kernel<<<(2 * B1N) / 8, 256, 0, stream>>>(e1, e2, nrm);

    dim3 grid(B1N / BT, B1N / BT, 2);   // (64, 64, 2)
    cl_gram_exp_kernel<<<grid, 256, 0, stream>>>(e1, e2, nrm, accum);

    cl_final_kernel<<<1, 32, 0, stream>>>(accum, out);
}